// magnet_loss_14319420965166
// MI455X (gfx1250) — compile-verified
//
#include <hip/hip_runtime.h>
#include <math.h>

#define B_N   16384
#define M_N   512
#define D_N   32
#define DIM_N 1024
#define EPS_F 1e-12f
#define ALPHA_F 7.18f

#define KC    8           // K-chunk staged in LDS (double buffered)
#define PITCH 12          // LDS row pitch in floats (8 data + 4 pad; stride 12 mod 64 -> 16 distinct banks)
#define NCHUNK (DIM_N / KC)

// single shared array -> LDS offsets are plain byte offsets from 0 (group segment base)
#define BUF_FLOATS (M_N * PITCH + 32 * PITCH)     // 6528 floats per buffer (Bs then As)
#define AS_BASE    (M_N * PITCH)                  // As offset within a buffer
#define PSUM_OFF   (2 * BUF_FLOATS)               // float psum[4][32]
#define OWNE_OFF   (PSUM_OFF + 128)               // float own_e[32]
#define OWND_OFF   (OWNE_OFF + 32)                // float own_d[32]
#define SMEM_FLOATS (OWND_OFF + 32)               // 13248 floats = 52992 B

typedef __attribute__((ext_vector_type(2))) float v2f;
typedef __attribute__((ext_vector_type(8))) float v8f;

// ---- workspace layout (bytes) ----
#define WS_LABELS   0                           // int[B]
#define WS_MEMBERS  (WS_LABELS  + 4*B_N)        // int[B]  (members[m*32+k])
#define WS_MEANS    (WS_MEMBERS + 4*B_N)        // float[M*DIM]
#define WS_C2       (WS_MEANS   + 4*M_N*DIM_N)  // float[M]
#define WS_X2       (WS_C2      + 4*M_N)        // float[B]
#define WS_SDIST    (WS_X2      + 4*B_N)        // float[B]
#define WS_LOSSI    (WS_SDIST   + 4*B_N)        // float[B]
#define WS_STDEV    (WS_LOSSI   + 4*B_N)        // float[1]

// CDNA5 async global->LDS (ASYNCcnt path); LDS address is a byte offset VGPR.
__device__ __forceinline__ void async_load_b128(unsigned lds_off, const float* gaddr) {
    asm volatile("global_load_async_to_lds_b128 %0, %1, off"
                 :
                 : "v"(lds_off), "v"((unsigned long long)(size_t)gaddr)
                 : "memory");
}
__device__ __forceinline__ void wait_async0() {
    asm volatile("s_wait_asynccnt 0x0" ::: "memory");
}

// K0: labels[i] = assignment[indices[i]]
__global__ __launch_bounds__(256) void k0_labels(const int* __restrict__ indices,
                                                 const int* __restrict__ assignment,
                                                 int* __restrict__ labels) {
    int i = blockIdx.x * blockDim.x + threadIdx.x;
    if (i < B_N) labels[i] = assignment[indices[i]];
}

// K1: deterministic ordered member list per cluster (1 wave32 per cluster)
__global__ __launch_bounds__(32) void k1_members(const int* __restrict__ labels,
                                                 int* __restrict__ members) {
    const int m    = blockIdx.x;
    const int lane = threadIdx.x;
    int base = 0;
    for (int c = 0; c < B_N / 32; ++c) {
        int  i    = c * 32 + lane;
        bool pred = (labels[i] == m);
        unsigned mm = (unsigned)__ballot(pred);           // wave32: low 32 bits
        int before  = __popc(mm & ((1u << lane) - 1u));
        if (pred) members[m * D_N + base + before] = i;
        base += __popc(mm);
    }
}

// K2: cluster means (fixed-order 32-member sum) + c2[m] = ||mean||^2
__global__ __launch_bounds__(256) void k2_means(const float* __restrict__ outputs,
                                                const int* __restrict__ members,
                                                float* __restrict__ means,
                                                float* __restrict__ c2) {
    __shared__ int   midx[D_N];
    __shared__ float red[256];
    const int m   = blockIdx.x;
    const int tid = threadIdx.x;
    if (tid < D_N) midx[tid] = members[m * D_N + tid];
    __syncthreads();
    float sq = 0.0f;
#pragma unroll
    for (int j = 0; j < DIM_N / 256; ++j) {
        int col = tid + j * 256;
        float s = 0.0f;
        for (int k = 0; k < D_N; ++k)
            s += outputs[(size_t)midx[k] * DIM_N + col];
        float mu = s * (1.0f / (float)D_N);
        means[(size_t)m * DIM_N + col] = mu;
        sq += mu * mu;
    }
    red[tid] = sq;
    __syncthreads();
    for (int s = 128; s > 0; s >>= 1) { if (tid < s) red[tid] += red[tid + s]; __syncthreads(); }
    if (tid == 0) c2[m] = red[0];
}

// K3: per-instance x2 and sqrt(max(||x-mu||^2, eps))
__global__ __launch_bounds__(256) void k3_x2_sdist(const float* __restrict__ outputs,
                                                   const float* __restrict__ means,
                                                   const int* __restrict__ labels,
                                                   float* __restrict__ x2,
                                                   float* __restrict__ sdist) {
    __shared__ float redA[256], redB[256];
    const int i   = blockIdx.x;
    const int tid = threadIdx.x;
    const int lab = labels[i];
    float sA = 0.0f, sB = 0.0f;
#pragma unroll
    for (int j = 0; j < DIM_N / 256; ++j) {
        int col  = tid + j * 256;
        float v  = outputs[(size_t)i * DIM_N + col];
        float mu = means[(size_t)lab * DIM_N + col];
        sA += v * v;
        float d = v - mu;
        sB += d * d;
    }
    redA[tid] = sA; redB[tid] = sB;
    __syncthreads();
    for (int s = 128; s > 0; s >>= 1) {
        if (tid < s) { redA[tid] += redA[tid + s]; redB[tid] += redB[tid + s]; }
        __syncthreads();
    }
    if (tid == 0) { x2[i] = redA[0]; sdist[i] = sqrtf(fmaxf(redB[0], EPS_F)); }
}

// K4: stdev = mean(sdist); also writes d_out[1025]
__global__ __launch_bounds__(256) void k4_stdev(const float* __restrict__ sdist,
                                                float* __restrict__ stdev_p,
                                                float* __restrict__ d_out) {
    __shared__ float red[256];
    const int tid = threadIdx.x;
    float s = 0.0f;
    for (int i = tid; i < B_N; i += 256) s += sdist[i];
    red[tid] = s;
    __syncthreads();
    for (int st = 128; st > 0; st >>= 1) { if (tid < st) red[tid] += red[tid + st]; __syncthreads(); }
    if (tid == 0) { float sv = red[0] * (1.0f / (float)B_N); stdev_p[0] = sv; d_out[1025] = sv; }
}

// K5: fused fp32 WMMA GEMM (outputs @ means^T) + epilogue -> loss_i[row]
// Block: 32 rows x 512 cols, 8 waves = (row_tile 0..1) x (col_group 0..3).
// Double-buffered K-chunks of 8, prefetched with CDNA5 async global->LDS
// (ASYNCcnt), consumed via conflict-free ds_load fragments + v_wmma.
__global__ __launch_bounds__(256) void k5_gemm_epilogue(const float* __restrict__ outputs,
                                                        const float* __restrict__ means,
                                                        const float* __restrict__ x2,
                                                        const float* __restrict__ c2,
                                                        const int* __restrict__ labels,
                                                        const float* __restrict__ stdev_p,
                                                        float* __restrict__ loss_i) {
    __shared__ float smem[SMEM_FLOATS];     // [buf0: Bs|As][buf1: Bs|As][psum][own_e][own_d]

    const int tid   = threadIdx.x;
    const int wave  = tid >> 5;
    const int lane  = tid & 31;
    const int rt    = wave >> 2;        // row tile 0..1
    const int cg    = wave & 3;         // col group 0..3
    const int lhalf = lane >> 4;        // 0: K/M low half, 1: K/M high half
    const int lm    = lane & 15;

    const int rowBase0 = blockIdx.x * 32;
    const int rowBase  = rowBase0 + rt * 16;
    const int colBase  = cg * 128;

    v8f acc[8] = {};

    // ---- async staging of one K-chunk into buffer `buf` ----
    // B: 512 cols x 8 K = 1024 float4 -> 4 per thread (32B-coalesced groups of 2 lanes)
    // A: 32 rows x 8 K = 64 float4 -> waves 0..1 (wave-uniform predicate)
    auto stage = [&](int k0, int buf) {
        const unsigned bufBase = (unsigned)(buf * BUF_FLOATS);
#pragma unroll
        for (int i = 0; i < 4; ++i) {
            int idx = tid + i * 256;
            int col = idx >> 1;
            int kq  = idx & 1;
            unsigned lds = (bufBase + (unsigned)(col * PITCH + kq * 4)) * 4u;
            async_load_b128(lds, means + (size_t)col * DIM_N + k0 + kq * 4);
        }
        if (tid < 64) {
            int row = tid >> 1;
            int kq  = tid & 1;
            unsigned lds = (bufBase + (unsigned)(AS_BASE + row * PITCH + kq * 4)) * 4u;
            async_load_b128(lds, outputs + (size_t)(rowBase0 + row) * DIM_N + k0 + kq * 4);
        }
    };

    stage(0, 0);   // prefetch chunk 0

    for (int c = 0; c < NCHUNK; ++c) {
        wait_async0();          // my batch c landed in LDS
        __syncthreads();        // everyone's batch c visible; buffer (c+1)&1 free to overwrite
        if (c + 1 < NCHUNK) stage((c + 1) * KC, (c + 1) & 1);   // prefetch next chunk

        const float* Bs = &smem[(c & 1) * BUF_FLOATS];
        const float* As = Bs + AS_BASE;
#pragma unroll
        for (int ks = 0; ks < KC; ks += 4) {
            // A frag (ISA 32-bit A 16x4): lane lm = M, VGPR pair = K (lo/hi half via lhalf)
            v2f a = *(const v2f*)(&As[(rt * 16 + lm) * PITCH + ks + 2 * lhalf]);
#pragma unroll
            for (int nt = 0; nt < 8; ++nt) {
                // B frag: B[K][N] = means[N][K]; lane lm = N
                v2f b = *(const v2f*)(&Bs[(colBase + nt * 16 + lm) * PITCH + ks + 2 * lhalf]);
                acc[nt] = __builtin_amdgcn_wmma_f32_16x16x4_f32(
                    false, a, false, b, (short)0, acc[nt], false, false);
            }
        }
    }

    const float sdv = stdev_p[0];
    const float sc  = -0.5f / sdv;

    float* psum  = &smem[PSUM_OFF];   // [cg*32 + j]
    float* own_e = &smem[OWNE_OFF];
    float* own_d = &smem[OWND_OFF];

    // per-lane row metadata: VGPR r of C/D maps to row = rowBase + r + 8*lhalf
    float xr[8]; int lr[8];
#pragma unroll
    for (int r = 0; r < 8; ++r) {
        int row = rowBase + r + 8 * lhalf;
        xr[r] = x2[row];
        lr[r] = labels[row];
    }

    float p[8];
#pragma unroll
    for (int r = 0; r < 8; ++r) p[r] = 0.0f;

#pragma unroll
    for (int nt = 0; nt < 8; ++nt) {
        const int   col = colBase + nt * 16 + lm;   // N = lane within half
        const float cc  = c2[col];
#pragma unroll
        for (int r = 0; r < 8; ++r) {
            float d2   = xr[r] + cc - 2.0f * acc[nt][r];
            float dist = sqrtf(fmaxf(d2, EPS_F));
            float e    = expf(sc * dist);
            p[r] += e;
            if (col == lr[r]) {                     // unique writer per row
                int j = rt * 16 + r + 8 * lhalf;
                own_e[j] = e;
                own_d[j] = dist;
            }
        }
    }

    // reduce e-sums over the 16 lanes of each half (rows r and r+8)
#pragma unroll
    for (int r = 0; r < 8; ++r) {
        float v = p[r];
        v += __shfl_xor(v, 1, 16);
        v += __shfl_xor(v, 2, 16);
        v += __shfl_xor(v, 4, 16);
        v += __shfl_xor(v, 8, 16);
        if (lm == 0) psum[cg * 32 + rt * 16 + r + 8 * lhalf] = v;
    }
    __syncthreads();

    if (tid < 32) {
        int   row   = rowBase0 + tid;
        float tot   = psum[0 * 32 + tid] + psum[1 * 32 + tid] +
                      psum[2 * 32 + tid] + psum[3 * 32 + tid];
        float denom = tot - own_e[tid];
        float li    = fmaxf(logf(denom) + 0.5f * own_d[tid] / sdv + ALPHA_F, 0.0f);
        loss_i[row] = li;
    }
}

// K6: loss_vector/count outputs (fixed member order => deterministic)
__global__ __launch_bounds__(256) void k6_vec(const float* __restrict__ loss_i,
                                              const int* __restrict__ members,
                                              const float* __restrict__ lv_in,
                                              const float* __restrict__ lc_in,
                                              float* __restrict__ d_out) {
    int m = blockIdx.x * blockDim.x + threadIdx.x;
    if (m < M_N) {
        float s = 0.0f;
        for (int k = 0; k < D_N; ++k) s += loss_i[members[m * D_N + k]];
        d_out[1 + m]       = lv_in[m] + s;
        d_out[1 + M_N + m] = lc_in[m] + (float)D_N;
    }
}

// K7: scalar loss
__global__ __launch_bounds__(256) void k7_loss(const float* __restrict__ loss_i,
                                               float* __restrict__ d_out) {
    __shared__ float red[256];
    const int tid = threadIdx.x;
    float s = 0.0f;
    for (int i = tid; i < B_N; i += 256) s += loss_i[i];
    red[tid] = s;
    __syncthreads();
    for (int st = 128; st > 0; st >>= 1) { if (tid < st) red[tid] += red[tid + st]; __syncthreads(); }
    if (tid == 0) d_out[0] = red[0] * (1.0f / (float)B_N);
}

extern "C" void kernel_launch(void* const* d_in, const int* in_sizes, int n_in,
                              void* d_out, int out_size, void* d_ws, size_t ws_size,
                              hipStream_t stream) {
    const float* outputs    = (const float*)d_in[0];
    const int*   indices    = (const int*)d_in[1];
    const int*   assignment = (const int*)d_in[2];
    const float* lv_in      = (const float*)d_in[3];
    const float* lc_in      = (const float*)d_in[4];
    float* out = (float*)d_out;

    char* ws = (char*)d_ws;
    int*   labels  = (int*)(ws + WS_LABELS);
    int*   members = (int*)(ws + WS_MEMBERS);
    float* means   = (float*)(ws + WS_MEANS);
    float* c2      = (float*)(ws + WS_C2);
    float* x2      = (float*)(ws + WS_X2);
    float* sdist   = (float*)(ws + WS_SDIST);
    float* loss_i  = (float*)(ws + WS_LOSSI);
    float* stdev_p = (float*)(ws + WS_STDEV);

    k0_labels<<<B_N / 256, 256, 0, stream>>>(indices, assignment, labels);
    k1_members<<<M_N, 32, 0, stream>>>(labels, members);
    k2_means<<<M_N, 256, 0, stream>>>(outputs, members, means, c2);
    k3_x2_sdist<<<B_N, 256, 0, stream>>>(outputs, means, labels, x2, sdist);
    k4_stdev<<<1, 256, 0, stream>>>(sdist, stdev_p, out);
    k5_gemm_epilogue<<<B_N / 32, 256, 0, stream>>>(outputs, means, x2, c2, labels, stdev_p, loss_i);
    k6_vec<<<2, 256, 0, stream>>>(loss_i, members, lv_in, lc_in, out);
    k7_loss<<<1, 256, 0, stream>>>(loss_i, out);
}